// FeatureFusionLayer_2456721293611
// MI455X (gfx1250) — compile-verified
//
#include <hip/hip_runtime.h>
#include <hip/hip_bf16.h>
#include <math.h>
#include <cfloat>

typedef __attribute__((ext_vector_type(2))) float v2f;
typedef __attribute__((ext_vector_type(8))) float v8f;

#define NWIN 10752          // B*R*Tp = 32*3*112
#define NF 56
#define NK 7
#define NWF (NWIN*NF)       // 602112 windows*features
#define TSEQ 1176           // 3*56*7
#define NG 168              // 3*H (logical N)
#define NPAD 192            // padded N: 12 tiles of 16, 6 pairs of 32
#define NPAIRS 6
#define NBT 3584            // B*112 rows
#define HID 56

__device__ __forceinline__ void cswap(float &a, float &b) {
    float lo = fminf(a, b), hi = fmaxf(a, b);
    a = lo; b = hi;
}

// 19-comparator optimal sorting network for 8 elements
__device__ __forceinline__ void sort8(float* s) {
    cswap(s[0],s[1]); cswap(s[2],s[3]); cswap(s[4],s[5]); cswap(s[6],s[7]);
    cswap(s[0],s[2]); cswap(s[1],s[3]); cswap(s[4],s[6]); cswap(s[5],s[7]);
    cswap(s[1],s[2]); cswap(s[5],s[6]); cswap(s[0],s[4]); cswap(s[3],s[7]);
    cswap(s[1],s[5]); cswap(s[2],s[6]);
    cswap(s[1],s[4]); cswap(s[3],s[6]);
    cswap(s[2],s[4]); cswap(s[3],s[5]);
    cswap(s[3],s[4]);
}

__global__ void k_init(float* acc) {
    if (threadIdx.x < 8) acc[threadIdx.x] = 0.f;
}

// ---- zero-padded copy of a (rows,K) row-major matrix into (padRows,K)
__global__ void k_pad_b(const float* __restrict__ src, float* __restrict__ dst,
                        int rows, int K, int padRows) {
    int id = blockIdx.x * blockDim.x + threadIdx.x;
    if (id >= padRows * K) return;
    int r = id / K;
    dst[id] = (r < rows) ? src[id] : 0.f;   // same layout => same flat index when in range
}

// ---- window feature extraction: feat[(win,f),k], k in {max,min,mean,std(ddof1),skew,kurt,mad}
__global__ void k_feat(const float* __restrict__ x, float* __restrict__ feat) {
    int id = blockIdx.x * blockDim.x + threadIdx.x;
    if (id >= NWF) return;
    int f = id % NF;
    int m = id / NF;           // (b*R+r)*112 + tp
    int br = m / 112;
    int tp = m % 112;
    const float* p = x + ((size_t)br * 896 + (size_t)tp * 8) * NF + f;
    float v[8];
    #pragma unroll
    for (int j = 0; j < 8; ++j) v[j] = p[j * NF];
    float mx = v[0], mn = v[0], sum = v[0];
    #pragma unroll
    for (int j = 1; j < 8; ++j) { mx = fmaxf(mx, v[j]); mn = fminf(mn, v[j]); sum += v[j]; }
    float mean = sum * 0.125f;
    float d[8]; float ss = 0.f;
    #pragma unroll
    for (int j = 0; j < 8; ++j) { d[j] = v[j] - mean; ss += d[j] * d[j]; }
    float stds = sqrtf(ss * (1.0f / 7.0f));   // ddof=1
    float inv  = 1.0f / sqrtf(ss * 0.125f);   // 1/population-std
    float s2 = 0.f, s3 = 0.f, s4 = 0.f;
    #pragma unroll
    for (int j = 0; j < 8; ++j) {
        float z = d[j] * inv;
        float z2 = z * z;
        s2 += z2; s3 += z2 * z; s4 += z2 * z2;
    }
    float m2   = s2 * 0.125f;
    float skew = (s3 * 0.125f) / powf(m2, 1.5f);
    float kurt = (s4 * 0.125f) / (m2 * m2) - 3.0f;
    float srt[8];
    #pragma unroll
    for (int j = 0; j < 8; ++j) srt[j] = v[j];
    sort8(srt);
    float med = srt[3];
    float ad[8];
    #pragma unroll
    for (int j = 0; j < 8; ++j) ad[j] = fabsf(v[j] - med);
    sort8(ad);
    float mad = ad[3];
    float* o = feat + (size_t)id * NK;
    o[0] = mx; o[1] = mn; o[2] = mean; o[3] = stds; o[4] = skew; o[5] = kurt; o[6] = mad;
}

// ---- ReliefF: one block per group n; stable top-20 by (dist, index); accumulate 7 scores
__global__ void k_relieff(const float* __restrict__ feat, float* __restrict__ acc) {
    __shared__ float Xs[NF * NK];
    __shared__ float sq[NF];
    __shared__ float lacc[NK];
    int n = blockIdx.x;
    int tid = threadIdx.x;   // 64 threads
    for (int i = tid; i < NF * NK; i += blockDim.x)
        Xs[i] = feat[(size_t)n * (NF * NK) + i];
    if (tid < NK) lacc[tid] = 0.f;
    __syncthreads();
    if (tid < NF) {
        float s = 0.f;
        #pragma unroll
        for (int k = 0; k < NK; ++k) { float t = Xs[tid * NK + k]; s += t * t; }
        sq[tid] = s;
    }
    __syncthreads();
    if (tid < NF) {
        int d = tid;
        float myX[NK];
        #pragma unroll
        for (int k = 0; k < NK; ++k) myX[k] = Xs[d * NK + k];
        float sqd = sq[d];
        float bd[20]; int be[20];
        #pragma unroll
        for (int t = 0; t < 20; ++t) { bd[t] = FLT_MAX; be[t] = 1 << 30; }
        for (int e = 0; e < NF; ++e) {
            float dot = 0.f;
            #pragma unroll
            for (int k = 0; k < NK; ++k) dot += Xs[e * NK + k] * myX[k];
            float dv = sqd + sq[e] - 2.0f * dot;   // exact reference distance formula
            if (dv < bd[19] || (dv == bd[19] && e < be[19])) {
                int p = 19;
                while (p > 0 && (dv < bd[p-1] || (dv == bd[p-1] && e < be[p-1]))) {
                    bd[p] = bd[p-1]; be[p] = be[p-1]; --p;
                }
                bd[p] = dv; be[p] = e;
            }
        }
        float contrib[NK];
        #pragma unroll
        for (int k = 0; k < NK; ++k) contrib[k] = 0.f;
        for (int t = 0; t < 20; ++t) {
            int e = be[t];
            float sgn = (t < 10) ? -1.0f : 1.0f;   // hits subtract, misses add
            #pragma unroll
            for (int k = 0; k < NK; ++k)
                contrib[k] += sgn * fabsf(Xs[e * NK + k] - myX[k]);
        }
        #pragma unroll
        for (int k = 0; k < NK; ++k) atomicAdd(&lacc[k], contrib[k]);
    }
    __syncthreads();
    if (tid < NK) atomicAdd(&acc[tid], lacc[tid]);
}

// ---- 1x1 conv across the (reshaped) channel dim: seq[n, (o*56+w)*7+d]
__global__ void k_conv(const float* __restrict__ feat, const float* __restrict__ cw,
                       const float* __restrict__ cb, float* __restrict__ seq) {
    int id = blockIdx.x * blockDim.x + threadIdx.x;
    if (id >= NBT * TSEQ) return;
    int n   = id / TSEQ;
    int rem = id % TSEQ;
    int o   = rem / 392;
    int wd  = rem % 392;
    float s = cb[o];
    #pragma unroll
    for (int c = 0; c < 3; ++c)
        s += feat[((size_t)n * 3 + c) * 392 + wd] * cw[o * 3 + c];
    seq[id] = s;
}

// ---- f32 WMMA GEMM: C(M,N) = A(M,K) @ Bt^T + bias(N), Bt zero-padded to (NPAD,K).
// One wave computes a 16x32 tile (two accumulators share the A fragment).
// M % 16 == 0, K % 4 == 0. Inner loop: 3x b64 load + 2x wmma, no masking.
__global__ void k_gemm_wmma(const float* __restrict__ A, const float* __restrict__ Bt,
                            const float* __restrict__ bias, float* __restrict__ C,
                            int M, int N, int K) {
    int tm = blockIdx.x / NPAIRS;
    int tp = blockIdx.x % NPAIRS;
    int lane = threadIdx.x & 31;
    int half = lane >> 4;      // selects K-pair (A/B frag) and M-half (C frag)
    int l15  = lane & 15;
    int arow  = tm * 16 + l15;
    int bcol0 = tp * 32 + l15;
    int bcol1 = bcol0 + 16;
    const float* ap  = A  + (size_t)arow  * K + half * 2;
    const float* bp0 = Bt + (size_t)bcol0 * K + half * 2;
    const float* bp1 = Bt + (size_t)bcol1 * K + half * 2;
    v8f acc0 = {0.f, 0.f, 0.f, 0.f, 0.f, 0.f, 0.f, 0.f};
    v8f acc1 = {0.f, 0.f, 0.f, 0.f, 0.f, 0.f, 0.f, 0.f};
    for (int k0 = 0; k0 < K; k0 += 4) {
        v2f a, b0, b1;
        a.x  = ap[k0];   a.y  = ap[k0 + 1];
        b0.x = bp0[k0];  b0.y = bp0[k0 + 1];
        b1.x = bp1[k0];  b1.y = bp1[k0 + 1];
        acc0 = __builtin_amdgcn_wmma_f32_16x16x4_f32(false, a, false, b0,
                                                     (short)0, acc0, false, false);
        acc1 = __builtin_amdgcn_wmma_f32_16x16x4_f32(false, a, false, b1,
                                                     (short)0, acc1, false, false);
    }
    // C/D layout: VGPR v -> row = v + 8*half, col = l15 within tile
    if (bcol0 < N) {
        float bv = bias[bcol0];
        #pragma unroll
        for (int v = 0; v < 8; ++v) {
            int orow = tm * 16 + half * 8 + v;
            C[(size_t)orow * N + bcol0] = acc0[v] + bv;
        }
    }
    if (bcol1 < N) {
        float bv = bias[bcol1];
        #pragma unroll
        for (int v = 0; v < 8; ++v) {
            int orow = tm * 16 + half * 8 + v;
            C[(size_t)orow * N + bcol1] = acc1[v] + bv;
        }
    }
}

// ---- sequential GRU over 112 steps, one persistent workgroup; h,gh in LDS
__global__ void __launch_bounds__(1024) k_gru(const float* __restrict__ gx,
                                              const float* __restrict__ w_hh,
                                              const float* __restrict__ b_hh,
                                              float* __restrict__ hs) {
    __shared__ float h[32 * HID];     // 7 KB
    __shared__ float gh[32 * NG];     // 21.5 KB
    int tid = threadIdx.x;
    for (int i = tid; i < 32 * HID; i += 1024) h[i] = 0.f;
    __syncthreads();
    for (int t = 0; t < 112; ++t) {
        for (int o = tid; o < 32 * NG; o += 1024) {
            int b = o / NG, g = o % NG;
            const float* hr = &h[b * HID];
            const float* wr = &w_hh[g * HID];
            float s = 0.f;
            #pragma unroll 8
            for (int i = 0; i < HID; ++i) s += hr[i] * wr[i];
            gh[o] = s + b_hh[g];
        }
        __syncthreads();
        for (int o = tid; o < 32 * HID; o += 1024) {
            int b = o / HID, i = o % HID;
            const float* gxp = &gx[((size_t)b * 112 + t) * NG];
            float xr = gxp[i], xz = gxp[HID + i], xn = gxp[2 * HID + i];
            float hrv = gh[b * NG + i];
            float hzv = gh[b * NG + HID + i];
            float hnv = gh[b * NG + 2 * HID + i];
            float r  = 1.0f / (1.0f + expf(-(xr + hrv)));
            float z  = 1.0f / (1.0f + expf(-(xz + hzv)));
            float nn = tanhf(xn + r * hnv);
            float hv = (1.0f - z) * nn + z * h[o];
            h[o] = hv;
            hs[((size_t)b * 112 + t) * HID + i] = hv;
        }
        __syncthreads();
    }
}

// ---- fold scores into fc, writing zero-padded (NPAD,HID) B-matrix directly
__global__ void k_fold(const float* __restrict__ acc, const float* __restrict__ fc_w,
                       const float* __restrict__ fc_b, float* __restrict__ fw2,
                       float* __restrict__ fb2) {
    __shared__ float s[NK];
    int tid = threadIdx.x;   // 256 threads, 1 block
    if (tid < NK) s[tid] = acc[tid] * (1.0f / 188160.0f);  // nn*D*Tp*C = 10*56*112*3
    __syncthreads();
    for (int j = tid; j < NPAD; j += blockDim.x) {
        float b = 0.f;
        if (j < NG) {
            #pragma unroll
            for (int k = 0; k < NK; ++k) b += s[k] * fc_b[j * NK + k];
        }
        fb2[j] = b;
    }
    for (int idx = tid; idx < NPAD * HID; idx += blockDim.x) {
        int j = idx / HID, hcol = idx % HID;
        float w = 0.f;
        if (j < NG) {
            #pragma unroll
            for (int k = 0; k < NK; ++k) w += s[k] * fc_w[(j * NK + k) * HID + hcol];
        }
        fw2[idx] = w;
    }
}

extern "C" void kernel_launch(void* const* d_in, const int* in_sizes, int n_in,
                              void* d_out, int out_size, void* d_ws, size_t ws_size,
                              hipStream_t stream) {
    (void)in_sizes; (void)n_in; (void)out_size; (void)ws_size;
    const float* x      = (const float*)d_in[0];
    const float* conv_w = (const float*)d_in[1];
    const float* conv_b = (const float*)d_in[2];
    const float* w_ih   = (const float*)d_in[3];   // (168,1176) row-major
    const float* w_hh   = (const float*)d_in[4];
    const float* b_ih   = (const float*)d_in[5];
    const float* b_hh   = (const float*)d_in[6];
    const float* fc_w   = (const float*)d_in[7];
    const float* fc_b   = (const float*)d_in[8];
    float* out = (float*)d_out;                    // (3584,168) == (32,112,3,56)

    float* ws   = (float*)d_ws;
    float* feat = ws;                   // 4,214,784 floats
    float* seq  = feat + 4214784;       // 4,214,784
    float* gx   = seq  + 4214784;       //   602,112
    float* hs   = gx   + 602112;        //   200,704
    float* wihp = hs   + 200704;        //   225,792  (192 x 1176, zero-padded w_ih)
    float* fw2  = wihp + 225792;        //    10,752  (192 x 56, zero-padded folded fc_w)
    float* fb2  = fw2  + 10752;         //       192
    float* sacc = fb2  + 192;           //         8   (~38 MB total)

    k_init<<<1, 32, 0, stream>>>(sacc);
    k_pad_b<<<(NPAD * TSEQ + 255) / 256, 256, 0, stream>>>(w_ih, wihp, NG, TSEQ, NPAD);
    k_feat<<<(NWF + 255) / 256, 256, 0, stream>>>(x, feat);
    k_relieff<<<NWIN, 64, 0, stream>>>(feat, sacc);
    k_conv<<<(NBT * TSEQ + 255) / 256, 256, 0, stream>>>(feat, conv_w, conv_b, seq);

    int tilesM = NBT / 16;              // 224
    // GEMM1: gx = seq @ w_ih^T + b_ih   (M=3584, N=168, K=1176)
    k_gemm_wmma<<<tilesM * NPAIRS, 32, 0, stream>>>(seq, wihp, b_ih, gx, NBT, NG, TSEQ);

    k_gru<<<1, 1024, 0, stream>>>(gx, w_hh, b_hh, hs);
    k_fold<<<1, 256, 0, stream>>>(sacc, fc_w, fc_b, fw2, fb2);

    // GEMM2: out = hs @ fw2^T + fb2     (M=3584, N=168, K=56)
    k_gemm_wmma<<<tilesM * NPAIRS, 32, 0, stream>>>(hs, fw2, fb2, out, NBT, NG, HID);
}